// Interplot_15599321219570
// MI455X (gfx1250) — compile-verified
//
#include <hip/hip_runtime.h>
#include <stdint.h>

// Cell-parallel scatter-mean of Taylor-corrected node values (gfx1250).
//   out[c, ch] = mean_{e : cells_index[e]==c} ( phi[n] + r.g[n] + 0.5 r^T H[n] r ),
//   r = centroid[c] - mesh_pos[n],  n = cells_node[e],  cells_index sorted asc.
//
// Block = 256 consecutive cells. Two threads locate the block's contiguous
// entry window in the sorted index; the window's (cells_index, cells_node)
// slice is async-copied to LDS (global_load_async_to_lds_b32, ASYNCcnt);
// each thread then binary-searches ~10 steps in LDS instead of ~22 dependent
// global loads, and accumulates its 8 channels with b128 gathers (L2-resident:
// all node data = 116MB < 192MB L2). No atomics, no workspace, deterministic.

#define CCH 8
#define BLK 256
#define CAP 4096   // max entries per 256-cell window (mean ~768, sigma ~28)

__global__ __launch_bounds__(BLK)
void cell_taylor_mean_kernel(const float*  __restrict__ phi,    // [N,8]
                             const float*  __restrict__ grad,   // [N,8,2]
                             const float*  __restrict__ hess,   // [N,8,2,2]
                             const float2* __restrict__ pos,    // [N]
                             const float2* __restrict__ cent,   // [M]
                             const int*    __restrict__ cnode,  // [E]
                             const int*    __restrict__ cidx,   // [E] sorted asc
                             float*        __restrict__ out,    // [M,8]
                             int M, int E)
{
    __shared__ int    s_ci[CAP];     // staged cells_index window
    __shared__ int    s_cn[CAP];     // staged cells_node window
    __shared__ float2 s_cent[BLK];   // staged centroids
    __shared__ int    s_range[2];    // [wlo, whi)

    const int tid = threadIdx.x;
    const int c0  = blockIdx.x * BLK;
    const int c   = c0 + tid;
    const int cc  = (c < M) ? c : (M - 1);   // clamp tail lanes (uniform flow)

    // Async-stage this lane's centroid (8B) straight into LDS; waited on later.
    {
        unsigned lds_off = (unsigned)(uintptr_t)(&s_cent[tid]);
        const float2* gsrc = cent + cc;
        asm volatile("global_load_async_to_lds_b64 %0, %1, off"
                     :: "v"(lds_off), "v"(gsrc) : "memory");
    }

    // Two threads find the block's entry window: lower_bound(c0), lower_bound(c0+BLK).
    if (tid < 2) {
        const int target = c0 + tid * BLK;
        int lo = 0, hi = E;
        while (lo < hi) {
            int mid = (lo + hi) >> 1;
            if (cidx[mid] < target) lo = mid + 1; else hi = mid;
        }
        s_range[tid] = lo;
    }
    __syncthreads();

    const int wlo = s_range[0];
    const int whi = s_range[1];
    const int W   = whi - wlo;       // block-uniform

    float acc[CCH];
#pragma unroll
    for (int k = 0; k < CCH; ++k) acc[k] = 0.0f;
    int cnt = 0;
    float2 ce;

    auto accumulate = [&](int n) {
        const float2 p = pos[n];
        const float rx = ce.x - p.x;
        const float ry = ce.y - p.y;
        const float a2 = 0.5f * rx * rx;
        const float b2 = 0.5f * rx * ry;
        const float d2 = 0.5f * ry * ry;

        const float4* __restrict__ ph = (const float4*)(phi  + (size_t)n * 8);
        const float4* __restrict__ gr = (const float4*)(grad + (size_t)n * 16);
        const float4* __restrict__ hh = (const float4*)(hess + (size_t)n * 32);

        const float4 p0 = ph[0], p1 = ph[1];
        const float pv[CCH] = {p0.x, p0.y, p0.z, p0.w, p1.x, p1.y, p1.z, p1.w};

        const float4 g0 = gr[0], g1 = gr[1], g2 = gr[2], g3 = gr[3];
        const float gx[CCH] = {g0.x, g0.z, g1.x, g1.z, g2.x, g2.z, g3.x, g3.z};
        const float gy[CCH] = {g0.y, g0.w, g1.y, g1.w, g2.y, g2.w, g3.y, g3.w};

#pragma unroll
        for (int k = 0; k < CCH; ++k) {
            const float4 h = hh[k];                  // Hxx,Hxy,Hyx,Hyy
            const float second = a2 * h.x + b2 * (h.y + h.z) + d2 * h.w;
            acc[k] += pv[k] + rx * gx[k] + ry * gy[k] + second;
        }
        ++cnt;
    };

    if (W <= CAP) {
        // Coalesced async copy of the entry window into LDS (ASYNCcnt-tracked).
        for (int j = tid; j < W; j += BLK) {
            unsigned o1 = (unsigned)(uintptr_t)(&s_ci[j]);
            const int* g1 = cidx + wlo + j;
            asm volatile("global_load_async_to_lds_b32 %0, %1, off"
                         :: "v"(o1), "v"(g1) : "memory");
            unsigned o2 = (unsigned)(uintptr_t)(&s_cn[j]);
            const int* g2 = cnode + wlo + j;
            asm volatile("global_load_async_to_lds_b32 %0, %1, off"
                         :: "v"(o2), "v"(g2) : "memory");
        }
        // Each wave drains its own async ops, then block-barrier for visibility.
        asm volatile("s_wait_asynccnt 0" ::: "memory");
        __syncthreads();
        ce = s_cent[tid];

        if (c < M) {
            int lo = 0, hi = W;                       // ~10-step LDS search
            while (lo < hi) {
                int mid = (lo + hi) >> 1;
                if (s_ci[mid] < c) lo = mid + 1; else hi = mid;
            }
            for (int i = lo; i < W && s_ci[i] == c; ++i)
                accumulate(s_cn[i]);
        }
    } else {
        // Pathological window: fall back to per-thread global binary search.
        asm volatile("s_wait_asynccnt 0" ::: "memory");
        ce = s_cent[tid];
        if (c < M) {
            int lo = wlo, hi = whi;
            while (lo < hi) {
                int mid = (lo + hi) >> 1;
                if (cidx[mid] < c) lo = mid + 1; else hi = mid;
            }
            __builtin_prefetch(cnode + lo, 0, 0);     // global_prefetch_b8
            for (int i = lo; i < E && cidx[i] == c; ++i)
                accumulate(cnode[i]);
        }
    }

    if (c < M) {
        const float denom = (cnt > 0) ? (float)cnt : 1.0f;
        const float inv = 1.0f / denom;
        float4 o0, o1;
        o0.x = acc[0] * inv; o0.y = acc[1] * inv; o0.z = acc[2] * inv; o0.w = acc[3] * inv;
        o1.x = acc[4] * inv; o1.y = acc[5] * inv; o1.z = acc[6] * inv; o1.w = acc[7] * inv;
        float4* __restrict__ op = (float4*)(out + (size_t)c * 8);
        op[0] = o0;
        op[1] = o1;
    }
}

extern "C" void kernel_launch(void* const* d_in, const int* in_sizes, int n_in,
                              void* d_out, int out_size, void* d_ws, size_t ws_size,
                              hipStream_t stream) {
    (void)n_in; (void)d_ws; (void)ws_size; (void)out_size;
    const float*  phi   = (const float*) d_in[0];  // [N,8]
    const float*  grad  = (const float*) d_in[1];  // [N,8,2]
    const float*  hess  = (const float*) d_in[2];  // [N,8,2,2]
    const float2* pos   = (const float2*)d_in[3];  // [N,2]
    const float2* cent  = (const float2*)d_in[4];  // [M,2]
    const int*    cnode = (const int*)   d_in[5];  // [E]
    const int*    cidx  = (const int*)   d_in[6];  // [E] sorted
    float*        out   = (float*)       d_out;    // [M,8]

    const int M = in_sizes[4] / 2;   // N_CELLS
    const int E = in_sizes[5];       // N_ENTRIES

    const int grid = (M + BLK - 1) / BLK;
    cell_taylor_mean_kernel<<<grid, BLK, 0, stream>>>(
        phi, grad, hess, pos, cent, cnode, cidx, out, M, E);
}